// VectorQuantizer_36301063586074
// MI455X (gfx1250) — compile-verified
//
#include <hip/hip_runtime.h>
#include <float.h>

typedef __attribute__((ext_vector_type(2))) float v2f;
typedef __attribute__((ext_vector_type(4))) float v4f;
typedef __attribute__((ext_vector_type(8))) float v8f;

#define N_CODES   16384
#define E_DIM     64
#define N_ROWS    8192          // 8 * 32 * 32
#define NZ        524288        // 8 * 64 * 32 * 32
#define NSEG      4
#define SEG_CODES (N_CODES / NSEG)   // 4096
#define SEG_CHUNK (SEG_CODES / 16)   // 256 chunks of 16 codes
#define CHUNK_BYTES (16 * E_DIM * 4) // 4096
#define BUF_FLOATS (16 * E_DIM + 16) // emb chunk + 16 enorm values

// ---- ws layout (float units) ----
#define WS_ZF     0                       // [8192][64]
#define WS_ENORM  (WS_ZF + NZ)            // [16384]
#define WS_PBEST  (WS_ENORM + N_CODES)    // [8192][NSEG]
#define WS_PIDX   (WS_PBEST + N_ROWS*NSEG)// [8192][NSEG] (int)
#define WS_IDX    (WS_PIDX + N_ROWS*NSEG) // [8192] (int)
#define WS_ACC    (WS_IDX + N_ROWS)       // [1] loss accumulator

// ---- d_out layout (float units) ----
#define OUT_ZQ     0            // 524288  [B,C,H,W]
#define OUT_LOSS   NZ           // 1
#define OUT_IDX    (NZ + 1)     // 8192
#define OUT_NUNIQ  (NZ + 1 + N_ROWS)
#define OUT_USAGE  (OUT_NUNIQ + 1)       // 16384
#define OUT_TOTAL  (OUT_USAGE + N_CODES) // 1

__device__ __forceinline__ void wait_async0() {
#if __has_builtin(__builtin_amdgcn_s_wait_asynccnt)
    __builtin_amdgcn_s_wait_asynccnt(0);
#else
    asm volatile("s_wait_asynccnt 0x0" ::: "memory");
#endif
}

// -------------------------------------------------------------------------
__global__ void vq_zero_kernel(float* out, float* acc) {
    int t = blockIdx.x * blockDim.x + threadIdx.x;
    if (t < N_CODES) out[OUT_USAGE + t] = 0.0f;
    if (t == 0) acc[0] = 0.0f;
}

// z [8,64,32,32] -> zf [8192][64]
__global__ void vq_transpose_kernel(const float* __restrict__ z,
                                    float* __restrict__ zf) {
    int t = blockIdx.x * blockDim.x + threadIdx.x;
    if (t >= NZ) return;
    int n = t >> 6;          // row
    int c = t & 63;
    int b = n >> 10;
    int hw = n & 1023;
    zf[t] = z[((b * 64 + c) << 10) + hw];
}

__global__ void vq_enorm_kernel(const float* __restrict__ emb,
                                float* __restrict__ enorm) {
    int k = blockIdx.x * blockDim.x + threadIdx.x;
    if (k >= N_CODES) return;
    const v4f* row = (const v4f*)(emb + k * E_DIM);
    float s = 0.0f;
#pragma unroll
    for (int i = 0; i < E_DIM / 4; ++i) {
        v4f v = row[i];
        s += v.x * v.x + v.y * v.y + v.z * v.z + v.w * v.w;
    }
    enorm[k] = s;
}

// -------------------------------------------------------------------------
// Main distance/argmin kernel. grid = (64, NSEG), block = 256 (8 waves).
// Each wave: 16-row A-tile (pre-scaled by -2), iterates SEG_CHUNK chunks of
// 16 codes. Both the emb chunk (4KB) and its 16 enorm values are staged into
// LDS via global_load_async_to_lds_{b128,b32} (ASYNCcnt), waited only after
// the WMMA chain. The accumulator is initialized to ||e||^2, so the 16-deep
// v_wmma_f32_16x16x4_f32 chain directly yields score = ||e||^2 - 2 z.e.
__global__ __launch_bounds__(256) void vq_argmin_kernel(
    const float* __restrict__ zf, const float* __restrict__ emb,
    const float* __restrict__ enorm,
    float* __restrict__ pbest, int* __restrict__ pidx) {

    __shared__ float lds[2][BUF_FLOATS];   // double-buffered chunk + enorm

    const int tid  = threadIdx.x;
    const int wave = tid >> 5;
    const int lane = tid & 31;
    const int lrow = lane & 15;    // N (col) slot for B/C; M row for A
    const int lhi  = lane >> 4;    // lane-half selector
    const int rowbase = blockIdx.x * 128 + wave * 16;
    const int segbase = blockIdx.y * SEG_CODES;

    // per-thread LDS byte offsets (low 32 bits of generic pointer to
    // __shared__ == LDS byte address)
    const unsigned lds_emb0 = (unsigned)(size_t)(&lds[0][0]) + tid * 16;
    const unsigned lds_emb1 = (unsigned)(size_t)(&lds[1][0]) + tid * 16;
    const unsigned lds_en0  = (unsigned)(size_t)(&lds[0][16 * E_DIM]) + tid * 4;
    const unsigned lds_en1  = (unsigned)(size_t)(&lds[1][16 * E_DIM]) + tid * 4;
    // chunk j is a contiguous 4KB block of emb
    const char* gbase  = (const char*)emb + (size_t)segbase * (E_DIM * 4) + tid * 16;
    const char* gnbase = (const char*)(enorm + segbase + tid);

    auto stage_async = [&](int j, int buf) {
        const char* g = gbase + (size_t)j * CHUNK_BYTES;
        unsigned le = buf ? lds_emb1 : lds_emb0;
        asm volatile("global_load_async_to_lds_b128 %0, %1, off"
                     :: "v"(le), "v"(g) : "memory");
        if (tid < 16) {  // async ops are EXEC-masked: lanes 0-15 move 64B
            const char* gn = gnbase + (size_t)j * 64;
            unsigned ln = buf ? lds_en1 : lds_en0;
            asm volatile("global_load_async_to_lds_b32 %0, %1, off"
                         :: "v"(ln), "v"(gn) : "memory");
        }
    };

    // Load A tile pre-scaled by -2 (16 fragments of 16x4 fp32; lane = M,
    // K pair split across lane halves per ISA layout)
    v2f a[16];
    {
        const float* arow = zf + (rowbase + lrow) * E_DIM + lhi * 2;
#pragma unroll
        for (int kk = 0; kk < 16; ++kk) {
            v2f v = *(const v2f*)(arow + 4 * kk);
            a[kk].x = -2.0f * v.x;
            a[kk].y = -2.0f * v.y;
        }
    }

    stage_async(0, 0);
    wait_async0();
    __syncthreads();

    float best[8];
    int   bidx[8];
#pragma unroll
    for (int i = 0; i < 8; ++i) { best[i] = FLT_MAX; bidx[i] = 0; }

    for (int j = 0; j < SEG_CHUNK; ++j) {
        int cur = j & 1;
        if (j + 1 < SEG_CHUNK) stage_async(j + 1, cur ^ 1);

        float en = lds[cur][16 * E_DIM + lrow];      // ds_load_b32
        v8f c = {en, en, en, en, en, en, en, en};    // C init = ||e||^2
        const float* lp = &lds[cur][lrow * E_DIM + lhi * 2];
#pragma unroll
        for (int kk = 0; kk < 16; ++kk) {
            v2f b = *(const v2f*)(lp + 4 * kk);
            c = __builtin_amdgcn_wmma_f32_16x16x4_f32(
                    false, a[kk], false, b, (short)0, c, false, false);
        }

        int col = segbase + j * 16 + lrow;
#pragma unroll
        for (int i = 0; i < 8; ++i) {
            // c[i] = ||e||^2 - 2 z.e  (A was pre-scaled by -2)
            if (c[i] < best[i]) { best[i] = c[i]; bidx[i] = col; }
        }

        wait_async0();       // next buffer resident before anyone reads it
        __syncthreads();     // and all waves past reads of the buffer we refill
    }

    // cross-lane argmin over the 16 column slots (within each lane half)
#pragma unroll
    for (int i = 0; i < 8; ++i) {
        float bv = best[i];
        int   bi = bidx[i];
#pragma unroll
        for (int m = 1; m < 16; m <<= 1) {
            float ov = __shfl_xor(bv, m, 32);
            int   oi = __shfl_xor(bi, m, 32);
            if (ov < bv || (ov == bv && oi < bi)) { bv = ov; bi = oi; }
        }
        best[i] = bv; bidx[i] = bi;
    }
    if (lrow == 0) {
#pragma unroll
        for (int i = 0; i < 8; ++i) {
            int r = rowbase + lhi * 8 + i;
            pbest[r * NSEG + blockIdx.y] = best[i];
            pidx [r * NSEG + blockIdx.y] = bidx[i];
        }
    }
}

// -------------------------------------------------------------------------
__global__ void vq_combine_kernel(const float* __restrict__ pbest,
                                  const int* __restrict__ pidx,
                                  int* __restrict__ idx_i,
                                  float* __restrict__ out) {
    int n = blockIdx.x * blockDim.x + threadIdx.x;
    if (n >= N_ROWS) return;
    float bv = FLT_MAX; int bi = 0;
#pragma unroll
    for (int s = 0; s < NSEG; ++s) {
        float v = pbest[n * NSEG + s];
        int   i = pidx [n * NSEG + s];
        if (v < bv || (v == bv && i < bi)) { bv = v; bi = i; }
    }
    idx_i[n] = bi;
    out[OUT_IDX + n] = (float)bi;
    out[OUT_USAGE + bi] = 1.0f;     // races write identical value
}

// gather z_q (back to [B,C,H,W]) + fused loss partial sums
__global__ __launch_bounds__(256) void vq_gather_loss_kernel(
    const float* __restrict__ z, const float* __restrict__ emb,
    const int* __restrict__ idx_i, float* __restrict__ out,
    float* __restrict__ acc) {
    __shared__ float red[256];
    int t = blockIdx.x * blockDim.x + threadIdx.x;
    float d2 = 0.0f;
    if (t < NZ) {
        int bc = t >> 10;
        int hw = t & 1023;
        int b = bc >> 6;
        int c = bc & 63;
        int n = (b << 10) + hw;
        float e = emb[idx_i[n] * E_DIM + c];
        out[OUT_ZQ + t] = e;
        float diff = e - z[t];
        d2 = diff * diff;
    }
    red[threadIdx.x] = d2;
    __syncthreads();
    for (int s = 128; s > 0; s >>= 1) {
        if (threadIdx.x < s) red[threadIdx.x] += red[threadIdx.x + s];
        __syncthreads();
    }
    if (threadIdx.x == 0) atomicAdd(acc, red[0]);
}

__global__ __launch_bounds__(256) void vq_finalize_kernel(
    float* __restrict__ out, const float* __restrict__ acc) {
    __shared__ float red[256];
    float s = 0.0f;
    for (int i = threadIdx.x; i < N_CODES; i += 256) s += out[OUT_USAGE + i];
    red[threadIdx.x] = s;
    __syncthreads();
    for (int st = 128; st > 0; st >>= 1) {
        if (threadIdx.x < st) red[threadIdx.x] += red[threadIdx.x + st];
        __syncthreads();
    }
    if (threadIdx.x == 0) {
        float usum = red[0];
        out[OUT_NUNIQ] = usum;                       // num_unique
        out[OUT_TOTAL] = usum / (float)N_CODES;      // total_usage
        out[OUT_LOSS]  = (1.0f + 0.25f) * acc[0] / (float)NZ;
    }
}

// -------------------------------------------------------------------------
extern "C" void kernel_launch(void* const* d_in, const int* in_sizes, int n_in,
                              void* d_out, int out_size, void* d_ws, size_t ws_size,
                              hipStream_t stream) {
    const float* z   = (const float*)d_in[0];   // [8,64,32,32]
    const float* emb = (const float*)d_in[1];   // [16384,64]
    float* out = (float*)d_out;
    float* ws  = (float*)d_ws;

    float* zf    = ws + WS_ZF;
    float* enorm = ws + WS_ENORM;
    float* pbest = ws + WS_PBEST;
    int*   pidx  = (int*)(ws + WS_PIDX);
    int*   idx_i = (int*)(ws + WS_IDX);
    float* acc   = ws + WS_ACC;

    vq_zero_kernel<<<(N_CODES + 255) / 256, 256, 0, stream>>>(out, acc);
    vq_transpose_kernel<<<NZ / 256, 256, 0, stream>>>(z, zf);
    vq_enorm_kernel<<<N_CODES / 256, 256, 0, stream>>>(emb, enorm);

    dim3 grid(64, NSEG);
    vq_argmin_kernel<<<grid, 256, 0, stream>>>(zf, emb, enorm, pbest, pidx);

    vq_combine_kernel<<<N_ROWS / 256, 256, 0, stream>>>(pbest, pidx, idx_i, out);
    vq_gather_loss_kernel<<<NZ / 256, 256, 0, stream>>>(z, emb, idx_i, out, acc);
    vq_finalize_kernel<<<1, 256, 0, stream>>>(out, acc);
}